// Router_2645699854601
// MI455X (gfx1250) — compile-verified
//
#include <hip/hip_runtime.h>
#include <hip/hip_bf16.h>

typedef float v2f __attribute__((ext_vector_type(2)));
typedef float v8f __attribute__((ext_vector_type(8)));

#define D_MODEL      4096
#define NUM_EXPERTS  64
#define TOKENS       16384
#define ROWS_PER_BLK 32
#define KC           64
#define AB_STRIDE    68   // 64 + 4 pad: keeps float4 (16B) alignment, bank stride 4
#define LG_STRIDE    65   // logits tile padding, conflict-free column scans

__global__ __launch_bounds__(256) void router_kernel(const float* __restrict__ x,
                                                     const float* __restrict__ W,
                                                     float* __restrict__ out) {
    __shared__ float Ash[ROWS_PER_BLK * AB_STRIDE];   // x tile   (32 x 64, padded)
    __shared__ float Bsh[NUM_EXPERTS * AB_STRIDE];    // W tile   (64 x 64, padded)
    __shared__ float Lsh[ROWS_PER_BLK * LG_STRIDE];   // logits   (32 x 64, padded)

    const int tid      = threadIdx.x;
    const int lane     = tid & 31;
    const int wave     = tid >> 5;
    const int halfLane = lane & 15;
    const int kOff     = (lane >> 4) * 2;        // lanes 0-15: K{0,1}; lanes 16-31: K{2,3}
    const int nBase    = (wave & 3) * 16;        // expert tile for this wave
    const int mBase    = (wave >> 2) * 16;       // row group for this wave
    const int rowBase  = blockIdx.x * ROWS_PER_BLK;

    // Two independent accumulation chains -> 2x WMMA ILP within a wave
    v8f acc0 = {};
    v8f acc1 = {};

    for (int k0 = 0; k0 < D_MODEL; k0 += KC) {
        // ---- cooperative stage: A = x[rowBase..+32, k0..+64]  (512 float4, 2/thread)
        #pragma unroll
        for (int i = 0; i < 2; ++i) {
            int idx = i * 256 + tid;             // 0..511
            int r   = idx >> 4;                  // row 0..31
            int c4  = idx & 15;                  // float4 column
            float4 v = *(const float4*)(x + (size_t)(rowBase + r) * D_MODEL + k0 + c4 * 4);
            *(float4*)(Ash + r * AB_STRIDE + c4 * 4) = v;
        }
        // ---- cooperative stage: B = W[0..64, k0..+64]  (1024 float4, 4/thread)
        #pragma unroll
        for (int i = 0; i < 4; ++i) {
            int idx = i * 256 + tid;             // 0..1023
            int e   = idx >> 4;                  // expert 0..63
            int c4  = idx & 15;
            float4 v = *(const float4*)(W + (size_t)e * D_MODEL + k0 + c4 * 4);
            *(float4*)(Bsh + e * AB_STRIDE + c4 * 4) = v;
        }
        __syncthreads();

        // prefetch the next x chunk while we compute (global_prefetch_b8)
        if (k0 + KC < D_MODEL) {
            int r = tid >> 3;                    // 0..31
            int c = (tid & 7) * 8;
            __builtin_prefetch(x + (size_t)(rowBase + r) * D_MODEL + (k0 + KC) + c, 0, 0);
        }

        // ---- 16 WMMA steps over this K chunk, alternating accumulator chains
        #pragma unroll
        for (int kk = 0; kk < KC; kk += 8) {
            const float* ap = Ash + (mBase + halfLane) * AB_STRIDE + kk + kOff;
            const float* bp = Bsh + (nBase + halfLane) * AB_STRIDE + kk + kOff;
            v2f a0, b0, a1, b1;
            a0.x = ap[0]; a0.y = ap[1];
            b0.x = bp[0]; b0.y = bp[1];
            a1.x = ap[4]; a1.y = ap[5];
            b1.x = bp[4]; b1.y = bp[5];
            // 8 args: (neg_a, A, neg_b, B, c_mod, C, reuse_a, reuse_b)
            acc0 = __builtin_amdgcn_wmma_f32_16x16x4_f32(
                false, a0, false, b0, (short)0, acc0, false, false);
            acc1 = __builtin_amdgcn_wmma_f32_16x16x4_f32(
                false, a1, false, b1, (short)0, acc1, false, false);
        }
        __syncthreads();
    }

    // ---- merge chains and stash 16x16 tiles into the shared logits buffer
    // C/D layout: VGPR r -> row (r + 8*(lane>>4)), col = lane&15
    #pragma unroll
    for (int r = 0; r < 8; ++r) {
        int row = mBase + r + 8 * (lane >> 4);
        Lsh[row * LG_STRIDE + nBase + halfLane] = acc0[r] + acc1[r];
    }
    __syncthreads();

    // ---- coalesced logits store: 32*64 = 2048 values, 8/thread
    float* logitsOut = out + 2 * TOKENS + 2 * TOKENS;   // after weights + indices
    #pragma unroll
    for (int i = 0; i < 8; ++i) {
        int idx = i * 256 + tid;                 // 0..2047
        int row = idx >> 6;
        int e   = idx & 63;
        logitsOut[(size_t)(rowBase + row) * NUM_EXPERTS + e] = Lsh[row * LG_STRIDE + e];
    }

    // ---- top-2 + renormalized softmax weights (denominator cancels)
    if (tid < ROWS_PER_BLK) {
        int row = tid;
        float l1 = -INFINITY, l2 = -INFINITY;
        int   i1 = 0, i2 = 0;
        #pragma unroll 4
        for (int e = 0; e < NUM_EXPERTS; ++e) {
            float v = Lsh[row * LG_STRIDE + e];
            if (v > l1)      { l2 = l1; i2 = i1; l1 = v; i1 = e; }
            else if (v > l2) { l2 = v;  i2 = e; }
        }
        float rr = __expf(l2 - l1);
        float w1 = 1.0f / (1.0f + rr);
        float w2 = 1.0f - w1;
        int t = rowBase + row;
        out[2 * t + 0] = w1;
        out[2 * t + 1] = w2;
        int* idxOut = (int*)out + 2 * TOKENS;
        idxOut[2 * t + 0] = i1;
        idxOut[2 * t + 1] = i2;
    }
}

extern "C" void kernel_launch(void* const* d_in, const int* in_sizes, int n_in,
                              void* d_out, int out_size, void* d_ws, size_t ws_size,
                              hipStream_t stream) {
    const float* x = (const float*)d_in[0];   // (4, 4096, 4096) f32
    const float* W = (const float*)d_in[1];   // (64, 4096) f32
    float* out = (float*)d_out;               // [weights 2T | indices 2T | logits 64T]
    (void)in_sizes; (void)n_in; (void)out_size; (void)d_ws; (void)ws_size;

    dim3 grid(TOKENS / ROWS_PER_BLK);         // 512 blocks
    dim3 block(256);                          // 8 waves: 2 M-groups x 4 N-tiles
    router_kernel<<<grid, block, 0, stream>>>(x, W, out);
}